// GCN_59356448031327
// MI455X (gfx1250) — compile-verified
//
#include <hip/hip_runtime.h>
#include <cstdint>
#include <cstddef>

// GCN forward for MI455X (gfx1250, wave32).
// - Feature matrices (51 MB) are L2-resident (192 MB L2) -> edge gather /
//   scatter phase is L2-bandwidth bound; GEMM phase is tiny (11.5 GFLOP).
// - Dense h@W via V_WMMA_F32_16X16X4_F32 (f32 keeps reference numerics;
//   compute is nowhere near the bottleneck so no reason to downcast).
// - Scatter-add via hardware GLOBAL_ATOMIC_ADD_F32 (inline asm, no CAS loop).
// - Self-loop (norm = dis^2) + bias fused into GEMM epilogue; ReLU fused
//   into the next layer's A-operand load (compile-time template flag).

typedef __attribute__((ext_vector_type(2))) float v2f;
typedef __attribute__((ext_vector_type(8))) float v8f;

#define HDIM 128

// ---------------------------------------------------------------- atomics ---
__device__ __forceinline__ void atomic_fadd(float* p, float v) {
  // No-return hardware f32 atomic (STOREcnt tracked; S_ENDPGM waits idle).
  asm volatile("global_atomic_add_f32 %0, %1, off" :: "v"(p), "v"(v) : "memory");
}

// ------------------------------------------------------------ small utils ---
__global__ void k_fill(float* __restrict__ p, float val, int n) {
  int i = blockIdx.x * blockDim.x + threadIdx.x;
  if (i < n) p[i] = val;
}

__global__ void k_deg_accum(const int* __restrict__ dst, float* __restrict__ deg, int E) {
  int i = blockIdx.x * blockDim.x + threadIdx.x;
  if (i < E) atomic_fadd(&deg[dst[i]], 1.0f);
}

__global__ void k_rsqrt_inplace(float* __restrict__ p, int n) {
  int i = blockIdx.x * blockDim.x + threadIdx.x;
  if (i < n) p[i] = rsqrtf(p[i]);   // deg >= 1 always (self-loop)
}

// ------------------------------------------------------------ WMMA GEMM -----
// OUT = act(A[n_rows,K]) @ W[K,128], n_rows % 32 == 0 (no guards needed).
// Block = 128 threads = 4 waves; wave w owns column tiles {2w, 2w+1}; every
// wave covers row tiles {0,1} of this block's 32 rows. 2x2 accumulator tile
// per wave: per K-step 2 A-loads (b64) + 4 B-loads (b32) feed 4 WMMAs.
//
// V_WMMA_F32_16X16X4_F32 operand layout (ISA 7.12.2):
//   A 16x4 : lane = M (mod 16); half-wave selects K pair {0,1} / {2,3}
//   B 4x16 : lane = N (mod 16); half-wave selects K pair (mirror of A)
//   C/D    : VGPR v, lane -> D[M = v + 8*half][N = lane&15]
template <int K, bool RELU>
__global__ void __launch_bounds__(128)
k_gemm_agg_init(const float* __restrict__ A,     // [n_rows, K]
                const float* __restrict__ W,     // [K, 128]
                const float* __restrict__ bias,  // [128]
                const float* __restrict__ dis,   // [n_rows]
                float* __restrict__ hw,          // [n_rows, 128]
                float* __restrict__ agg)         // [n_rows, 128]
{
  const int wave = threadIdx.x >> 5;           // 0..3
  const int lane = threadIdx.x & 31;
  const int r    = lane & 15;
  const int half = lane >> 4;                  // 0 or 1
  const int row0 = blockIdx.x * 32;            // row tiles: row0, row0+16
  const int c0   = wave * 32;                  // col tiles: c0, c0+16

  const float* A0 = A + (size_t)(row0 + r) * K + 2 * half;
  const float* A1 = A0 + (size_t)16 * K;
  const float* Wp = W + (size_t)2 * half * HDIM + r;

  v8f acc00 = {}, acc01 = {}, acc10 = {}, acc11 = {};
#pragma unroll 4
  for (int k = 0; k < K; k += 4) {
    v2f a0 = *reinterpret_cast<const v2f*>(A0 + k);
    v2f a1 = *reinterpret_cast<const v2f*>(A1 + k);
    if (RELU) {
      a0.x = fmaxf(a0.x, 0.0f); a0.y = fmaxf(a0.y, 0.0f);
      a1.x = fmaxf(a1.x, 0.0f); a1.y = fmaxf(a1.y, 0.0f);
    }
    const float* Wk = Wp + (size_t)k * HDIM;
    v2f b0, b1;
    b0.x = Wk[c0];            b0.y = Wk[HDIM + c0];
    b1.x = Wk[c0 + 16];       b1.y = Wk[HDIM + c0 + 16];
    acc00 = __builtin_amdgcn_wmma_f32_16x16x4_f32(false, a0, false, b0, (short)0, acc00, false, false);
    acc10 = __builtin_amdgcn_wmma_f32_16x16x4_f32(false, a1, false, b0, (short)0, acc10, false, false);
    acc01 = __builtin_amdgcn_wmma_f32_16x16x4_f32(false, a0, false, b1, (short)0, acc01, false, false);
    acc11 = __builtin_amdgcn_wmma_f32_16x16x4_f32(false, a1, false, b1, (short)0, acc11, false, false);
  }

  // Epilogue: hw = A@W ; agg = bias + dis^2 * hw   (self-loop fused)
  const int   colA = c0 + r, colB = c0 + 16 + r;
  const float bcA  = bias[colA], bcB = bias[colB];
#pragma unroll
  for (int v = 0; v < 8; ++v) {
    const int   m0  = row0 + v + 8 * half;
    const int   m1  = m0 + 16;
    const float d0  = dis[m0], d1 = dis[m1];
    const float q0  = d0 * d0, q1 = d1 * d1;
    float* h0 = hw  + (size_t)m0 * HDIM;
    float* h1 = hw  + (size_t)m1 * HDIM;
    float* g0 = agg + (size_t)m0 * HDIM;
    float* g1 = agg + (size_t)m1 * HDIM;
    h0[colA] = acc00[v];  g0[colA] = bcA + q0 * acc00[v];
    h0[colB] = acc01[v];  g0[colB] = bcB + q0 * acc01[v];
    h1[colA] = acc10[v];  g1[colA] = bcA + q1 * acc10[v];
    h1[colB] = acc11[v];  g1[colB] = bcB + q1 * acc11[v];
  }
}

// --------------------------------------------------------- edge scatter -----
// One wave per edge: lane handles 4 channels (float4 gather from L2-resident
// hw[src], 4 hardware f32 atomics into agg[dst]).
__global__ void __launch_bounds__(256)
k_edge_scatter(const int* __restrict__ src, const int* __restrict__ dst,
               const float* __restrict__ dis,
               const float* __restrict__ hw, float* __restrict__ agg, int E)
{
  const int e = (int)((blockIdx.x * (unsigned)blockDim.x + threadIdx.x) >> 5);
  if (e >= E) return;
  const int lane = threadIdx.x & 31;

  const int   s   = src[e];
  const int   d   = dst[e];
  const float nrm = dis[s] * dis[d];

  const float4 v = reinterpret_cast<const float4*>(hw + (size_t)s * HDIM)[lane];
  float* ap = agg + (size_t)d * HDIM + lane * 4;
  atomic_fadd(ap + 0, nrm * v.x);
  atomic_fadd(ap + 1, nrm * v.y);
  atomic_fadd(ap + 2, nrm * v.z);
  atomic_fadd(ap + 3, nrm * v.w);
}

// ----------------------------------------------------------------- pool -----
__global__ void __launch_bounds__(256)
k_pool_accum(const int* __restrict__ batch, const float* __restrict__ h,
             float* __restrict__ pooled, float* __restrict__ cnt, int n_rows)
{
  const int node = (int)((blockIdx.x * (unsigned)blockDim.x + threadIdx.x) >> 5);
  if (node >= n_rows) return;
  const int lane = threadIdx.x & 31;
  const int g    = batch[node];

  const float4 v = reinterpret_cast<const float4*>(h + (size_t)node * HDIM)[lane];
  float* pp = pooled + (size_t)g * HDIM + lane * 4;
  atomic_fadd(pp + 0, v.x);
  atomic_fadd(pp + 1, v.y);
  atomic_fadd(pp + 2, v.z);
  atomic_fadd(pp + 3, v.w);
  if (lane == 0) atomic_fadd(&cnt[g], 1.0f);
}

// ----------------------------------------------------------------- head -----
__global__ void k_head(const float* __restrict__ pooled, const float* __restrict__ cnt,
                       const float* __restrict__ Wlin, const float* __restrict__ blin,
                       float* __restrict__ out, int G, int C)
{
  const int idx = blockIdx.x * blockDim.x + threadIdx.x;
  if (idx >= G * C) return;
  const int g = idx / C;
  const int c = idx % C;
  const float inv = 1.0f / fmaxf(cnt[g], 1.0f);
  float s = blin[c];
  const float* pg = pooled + (size_t)g * HDIM;
#pragma unroll 4
  for (int k = 0; k < HDIM; ++k) s += pg[k] * inv * Wlin[(size_t)k * C + c];
  out[idx] = s;
}

// ---------------------------------------------------------------- launch ----
static inline size_t align256(size_t x) { return (x + 255) & ~(size_t)255; }

extern "C" void kernel_launch(void* const* d_in, const int* in_sizes, int n_in,
                              void* d_out, int out_size, void* d_ws, size_t ws_size,
                              hipStream_t stream)
{
  const int DIN = 64, C = 2;
  const int N = in_sizes[0] / DIN;       // 100000 (multiple of 32)
  const int E = in_sizes[1] / 2;         // 1600000
  const int G = out_size / C;            // 512

  const float* x    = (const float*)d_in[0];
  const int*   src  = (const int*)d_in[1];
  const int*   dst  = src + E;
  const int*   batch= (const int*)d_in[2];
  const float* W1 = (const float*)d_in[3],  *b1 = (const float*)d_in[4];
  const float* W2 = (const float*)d_in[5],  *b2 = (const float*)d_in[6];
  const float* W3 = (const float*)d_in[7],  *b3 = (const float*)d_in[8];
  const float* W4 = (const float*)d_in[9],  *b4 = (const float*)d_in[10];
  const float* Wlin = (const float*)d_in[11], *blin = (const float*)d_in[12];
  float* out = (float*)d_out;

  // Workspace carve-up (~154 MB; all L2-friendly)
  char* ws = (char*)d_ws;
  const size_t featB = (size_t)N * HDIM * sizeof(float);
  size_t off = 0;
  float* dis    = (float*)(ws + off); off = align256(off + (size_t)N * sizeof(float));
  float* hw     = (float*)(ws + off); off = align256(off + featB);
  float* aggA   = (float*)(ws + off); off = align256(off + featB);
  float* aggB   = (float*)(ws + off); off = align256(off + featB);
  float* pooled = (float*)(ws + off); off += (size_t)G * HDIM * sizeof(float);
  float* cnt    = (float*)(ws + off); off += (size_t)G * sizeof(float);
  (void)ws_size;

  const int TB = 256;
  const dim3 blk(TB);

  // Degree (init 1.0 for self-loop, +1 per incoming edge), then dis = rsqrt(deg)
  k_fill<<<dim3((N + TB - 1) / TB), blk, 0, stream>>>(dis, 1.0f, N);
  k_deg_accum<<<dim3((E + TB - 1) / TB), blk, 0, stream>>>(dst, dis, E);
  k_rsqrt_inplace<<<dim3((N + TB - 1) / TB), blk, 0, stream>>>(dis, N);

  const dim3 gemmBlk(128);                      // 4 waves
  const dim3 gemmGrid((unsigned)(N / 32));      // 3125 blocks, 32 rows each
  const dim3 edgeGrid((unsigned)((E + 7) / 8)); // 1 wave per edge
  const dim3 nodeGrid((unsigned)((N + 7) / 8)); // 1 wave per node

  // Layer 1: x[N,64] @ W1 -> hw ; aggA = b1 + dis^2*hw ; scatter edges
  k_gemm_agg_init<64, false><<<gemmGrid, gemmBlk, 0, stream>>>(x, W1, b1, dis, hw, aggA);
  k_edge_scatter<<<edgeGrid, blk, 0, stream>>>(src, dst, dis, hw, aggA, E);

  // Layer 2 (relu on input): aggA -> aggB
  k_gemm_agg_init<128, true><<<gemmGrid, gemmBlk, 0, stream>>>(aggA, W2, b2, dis, hw, aggB);
  k_edge_scatter<<<edgeGrid, blk, 0, stream>>>(src, dst, dis, hw, aggB, E);

  // Layer 3: aggB -> aggA
  k_gemm_agg_init<128, true><<<gemmGrid, gemmBlk, 0, stream>>>(aggB, W3, b3, dis, hw, aggA);
  k_edge_scatter<<<edgeGrid, blk, 0, stream>>>(src, dst, dis, hw, aggA, E);

  // Layer 4 (no relu on output): aggA -> aggB
  k_gemm_agg_init<128, true><<<gemmGrid, gemmBlk, 0, stream>>>(aggA, W4, b4, dis, hw, aggB);
  k_edge_scatter<<<edgeGrid, blk, 0, stream>>>(src, dst, dis, hw, aggB, E);

  // Global mean pool + linear head
  k_fill<<<dim3((G * (HDIM + 1) + TB - 1) / TB), blk, 0, stream>>>(pooled, 0.0f, G * (HDIM + 1));
  k_pool_accum<<<nodeGrid, blk, 0, stream>>>(batch, aggB, pooled, cnt, N);
  k_head<<<dim3((G * C + TB - 1) / TB), blk, 0, stream>>>(pooled, cnt, Wlin, blin, out, G, C);
}